// DeepBoundaryTree_24223615550372
// MI455X (gfx1250) — compile-verified
//
#include <hip/hip_runtime.h>
#include <math.h>

#define EPS_PD 1e-6f
#define EPS_LOG 1e-4f

typedef __attribute__((ext_vector_type(2))) float v2f;
typedef __attribute__((ext_vector_type(8))) float v8f;

static constexpr int Bq = 8192;
static constexpr int Nn = 100000;
static constexpr int Dd = 8;
static constexpr int Kk = 32;
static constexpr int Cc = 100;
static constexpr int TOTAL = Bq + Nn;

static constexpr int WAVES_A = 4;           // waves per block in MLP kernel
static constexpr int ROWS_PER_BLOCK = WAVES_A * 16;
static constexpr int HPAD = 112;            // 100 padded to 7 N-tiles of 16

// fp32 WMMA: D(16x16) = A(16x4) * B(4x16) + C ; exact fp32, matches reference math
__device__ __forceinline__ v8f wmma4(v2f a, v2f b, v8f c) {
    return __builtin_amdgcn_wmma_f32_16x16x4_f32(
        /*neg_a=*/false, a, /*neg_b=*/false, b,
        /*c_mod=*/(short)0, c, /*reuse_a=*/false, /*reuse_b=*/false);
}

// One MLP layer for a wave's 16-row tile, dimensions as template constants so
// the full-tile guards constant-fold away (no exec-mask churn in the K-loop).
//   out[M][t*16+n] = act( sum_k in[M][k] * W[n][k] + bias[n] ), zeros in padding.
// A-matrix layout (32-bit, 16x4): lanes 0-15 hold K=0(v0),K=1(v1); lanes 16-31 K=2,K=3.
// B-matrix layout mirrors: lanes 0-15 hold rows K=0/1, lanes 16-31 rows K=2/3, N=lane%16.
template <int O, int I, int KSTEPS, int NTILES, bool RELU>
__device__ __forceinline__ void mlp_layer(
    const float* __restrict__ W, const float* __restrict__ bias,
    const float (*__restrict__ bin)[HPAD], float (*__restrict__ bout)[HPAD],
    int m, bool hi)
{
#pragma unroll
    for (int t = 0; t < NTILES; ++t) {
        const int n = t * 16 + m;
        const bool nOK = (t * 16 + 16 <= O) ? true : (n < O);  // folds for full tiles
        const float bb = nOK ? bias[n] : 0.f;
        v8f acc = {};
        for (int ks = 0; ks < KSTEPS; ++ks) {
            const int k = ks * 4 + (hi ? 2 : 0);
            v2f a = *(const v2f*)&bin[m][k];          // padded cols are pre-zeroed
            v2f b = {0.f, 0.f};
            // valid K-pairs are whole pairs for I in {2,30,100}
            const bool kOK = (KSTEPS * 4 <= I) ? true : ((k + 1) < I);
            if (nOK && kOK) b = *(const v2f*)&W[n * I + k];
            acc = wmma4(a, b, acc);
        }
#pragma unroll
        for (int r = 0; r < 8; ++r) {
            const int M = r + (hi ? 8 : 0);
            float h = acc[r] + bb;
            h = nOK ? (RELU ? fmaxf(h, 0.f) : h) : 0.f;
            bout[M][t * 16 + m] = h;
        }
    }
}

__global__ void __launch_bounds__(WAVES_A * 32)
mlp_kernel(const float* __restrict__ x, const float* __restrict__ nd,
           const float* __restrict__ W1, const float* __restrict__ b1,
           const float* __restrict__ W2, const float* __restrict__ b2,
           const float* __restrict__ W3, const float* __restrict__ b3,
           const float* __restrict__ W4, const float* __restrict__ b4,
           float* __restrict__ qx, float* __restrict__ nx)
{
    __shared__ float bufA[WAVES_A][16][HPAD];
    __shared__ float bufB[WAVES_A][16][HPAD];

    const int wave = threadIdx.x >> 5;
    const int lane = threadIdx.x & 31;
    const int m = lane & 15;
    const bool hi = lane >= 16;
    const int rowBase = blockIdx.x * ROWS_PER_BLOCK + wave * 16;

    float (*bA)[HPAD] = bufA[wave];
    float (*bB)[HPAD] = bufB[wave];

    // Stage inputs [16][4] into bA cols 0..3 (cols 2,3 zero-padded K).
    for (int i = lane; i < 16 * 4; i += 32) {
        int r = i >> 2, c = i & 3;
        int grow = rowBase + r;
        float v = 0.f;
        if (c < 2 && grow < TOTAL)
            v = (grow < Bq) ? x[grow * 2 + c] : nd[(grow - Bq) * 2 + c];
        bA[r][c] = v;
    }
    __syncthreads();

    mlp_layer<100,   2,  1, 7, true>(W1, b1, bA, bB, m, hi);  // 2 -> 100
    __syncthreads();
    mlp_layer<100, 100, 25, 7, true>(W2, b2, bB, bA, m, hi);  // 100 -> 100
    __syncthreads();
    mlp_layer< 30, 100, 25, 2, true>(W3, b3, bA, bB, m, hi);  // 100 -> 30
    __syncthreads();

    // Layer 4: 30 -> 2, write straight to global (qx / nx).
    {
        v8f acc = {};
#pragma unroll
        for (int ks = 0; ks < 8; ++ks) {
            const int k = ks * 4 + (hi ? 2 : 0);
            v2f a = *(const v2f*)&bB[m][k];            // cols 30,31 are zero
            v2f b = {0.f, 0.f};
            if (m < 2 && (k + 1) < 30)
                b = *(const v2f*)&W4[m * 30 + k];
            acc = wmma4(a, b, acc);
        }
        if (m < 2) {
            const float bb = b4[m];
#pragma unroll
            for (int r = 0; r < 8; ++r) {
                const int M = r + (hi ? 8 : 0);
                const int grow = rowBase + M;
                if (grow < TOTAL) {
                    const float v = acc[r] + bb;
                    if (grow < Bq) qx[grow * 2 + m] = v;
                    else           nx[(grow - Bq) * 2 + m] = v;
                }
            }
        }
    }
}

// One block per query b; wave d (of 8) handles path depth d; lane k handles neighbor k.
__global__ void __launch_bounds__(Dd * 32)
route_kernel(const float* __restrict__ qx, const float* __restrict__ nx,
             const int* __restrict__ labels, const int* __restrict__ nbr,
             float* __restrict__ out)
{
    const int b = blockIdx.x;
    const int d = threadIdx.x >> 5;
    const int lane = threadIdx.x & 31;

    __shared__ float term1p[Dd];
    __shared__ float pcls[128];

    if (threadIdx.x < 128) pcls[threadIdx.x] = 0.f;
    __syncthreads();

    const int idx = nbr[(b * Dd + d) * Kk + lane];
    const float2 g = ((const float2*)nx)[idx];
    const float2 q = ((const float2*)qx)[b];
    const float dx = g.x - q.x + EPS_PD;
    const float dy = g.y - q.y + EPS_PD;
    const float v = -sqrtf(dx * dx + dy * dy);

    // wave32 butterfly reductions (K == warpSize == 32)
    float mx = v;
#pragma unroll
    for (int off = 16; off; off >>= 1) mx = fmaxf(mx, __shfl_xor(mx, off, 32));
    const float e = __expf(v - mx);
    float s = e;
#pragma unroll
    for (int off = 16; off; off >>= 1) s += __shfl_xor(s, off, 32);

    if (d < Dd - 1) {
        // max_k softmax = exp(vmax - mx)/s = 1/s
        if (lane == 0) term1p[d] = __logf(1.0f / s + EPS_LOG);
    } else {
        atomicAdd(&pcls[labels[idx]], e / s);  // ds_add_f32 class histogram
    }
    __syncthreads();

    float term1 = 0.f;
#pragma unroll
    for (int i = 0; i < Dd - 1; ++i) term1 += term1p[i];

    for (int c = threadIdx.x; c < Cc; c += Dd * 32)
        out[b * Cc + c] = __logf(pcls[c] + EPS_LOG) + term1;
}

extern "C" void kernel_launch(void* const* d_in, const int* in_sizes, int n_in,
                              void* d_out, int out_size, void* d_ws, size_t ws_size,
                              hipStream_t stream) {
    (void)in_sizes; (void)n_in; (void)out_size; (void)ws_size;
    const float* x      = (const float*)d_in[0];
    const float* nd     = (const float*)d_in[1];
    const float* W1     = (const float*)d_in[2];
    const float* b1     = (const float*)d_in[3];
    const float* W2     = (const float*)d_in[4];
    const float* b2     = (const float*)d_in[5];
    const float* W3     = (const float*)d_in[6];
    const float* b3     = (const float*)d_in[7];
    const float* W4     = (const float*)d_in[8];
    const float* b4     = (const float*)d_in[9];
    const int*   labels = (const int*)d_in[10];
    const int*   nbr    = (const int*)d_in[11];

    float* qx = (float*)d_ws;        // [B,2]
    float* nx = qx + Bq * 2;         // [N,2]
    float* out = (float*)d_out;      // [B,100]

    const int blocksA = (TOTAL + ROWS_PER_BLOCK - 1) / ROWS_PER_BLOCK;
    mlp_kernel<<<blocksA, WAVES_A * 32, 0, stream>>>(
        x, nd, W1, b1, W2, b2, W3, b3, W4, b4, qx, nx);
    route_kernel<<<Bq, Dd * 32, 0, stream>>>(qx, nx, labels, nbr, out);
}